// DocREModel_76716705841327
// MI455X (gfx1250) — compile-verified
//
#include <hip/hip_runtime.h>

// Problem dimensions (fixed by the reference)
#define B_   4
#define L_   1024
#define D_   768
#define H_   12
#define NE_  42
#define M_   8
#define NR_  (NE_ * NE_)   // 1764 entity pairs
#define OFFSET_ 1

// Kernel-3 tiling
#define KB_   64                 // K panel depth
#define NPAN_ (L_ / KB_)         // 16 panels
#define AS_   68                 // LDS row stride (floats) for A panel: 16 x KB
#define BS_   132                // LDS row stride (floats) for B panel: KB x 128
#define BN_   128                // block N tile (8 waves x 16 cols)

typedef __attribute__((ext_vector_type(2))) float v2f;
typedef __attribute__((ext_vector_type(8))) float v8f;

// ---------------------------------------------------------------------------
// CDNA5 async global->LDS helpers (VGLOBAL GLOBAL_LOAD_ASYNC_TO_LDS_B128,
// tracked by ASYNCcnt; see CDNA5 ISA).
// LDS destination address comes from the low 32 bits of the generic pointer
// (flat LDS aperture: addr[31:0] == LDS offset).
// ---------------------------------------------------------------------------
__device__ __forceinline__ uint32_t lds_off(const void* p) {
    return (uint32_t)(uintptr_t)p;
}

__device__ __forceinline__ void async_load_b128(uint32_t lds_addr, const float* gaddr) {
    asm volatile("global_load_async_to_lds_b128 %0, %1, off"
                 :: "v"(lds_addr), "v"(gaddr)
                 : "memory");
}

__device__ __forceinline__ void wait_async0() {
    asm volatile("s_wait_asynccnt 0x0" ::: "memory");
}

// ---------------------------------------------------------------------------
// Kernel 1: ent_att[b,h,e,l] = (1/cnt) * sum_m valid[m] * attention[b,h,idx[m],l]
// ---------------------------------------------------------------------------
__global__ void __launch_bounds__(256)
ent_att_kernel(const float* __restrict__ attention,
               const int*   __restrict__ mention_idx,
               const int*   __restrict__ mention_mask,
               float*       __restrict__ ent_att) {
    const int bid = blockIdx.x;            // b*H*NE + h*NE + e
    const int e   = bid % NE_;
    const int h   = (bid / NE_) % H_;
    const int b   = bid / (NE_ * H_);
    const int tid = threadIdx.x;

    int   idxs[M_];
    float vcnt = 0.f;
#pragma unroll
    for (int m = 0; m < M_; ++m) {
        const int base = (b * NE_ + e) * M_ + m;
        const int id   = mention_idx[base] + OFFSET_;
        const bool valid = (mention_mask[base] > 0) && (id < L_);
        idxs[m] = valid ? id : -1;
        vcnt   += valid ? 1.f : 0.f;
    }
    const float inv_cnt = 1.f / fmaxf(vcnt, 1.f);

    const float* att_bh = attention + (size_t)(b * H_ + h) * L_ * L_;
    float*       out    = ent_att   + (size_t)bid * L_;

    for (int l = tid; l < L_; l += 256) {
        float s = 0.f;
#pragma unroll
        for (int m = 0; m < M_; ++m)
            if (idxs[m] >= 0) s += att_bh[(size_t)idxs[m] * L_ + l];
        out[l] = s * inv_cnt;
    }
}

// ---------------------------------------------------------------------------
// Kernel 2: ht[b,s,t,l] = normalize_l( (1/H) * sum_h ent_att[s,l]*ent_att[t,l] )
// ---------------------------------------------------------------------------
__global__ void __launch_bounds__(256)
ht_att_kernel(const float* __restrict__ ent_att,
              float*       __restrict__ ht) {
    const int bid = blockIdx.x;            // b*NR + s*NE + t
    const int t   = bid % NE_;
    const int s   = (bid / NE_) % NE_;
    const int b   = bid / NR_;
    const int tid = threadIdx.x;

    const float* es = ent_att + (size_t)(b * H_ * NE_ + s) * L_;
    const float* et = ent_att + (size_t)(b * H_ * NE_ + t) * L_;
    const size_t hstride = (size_t)NE_ * L_;

    float vals[4];
    float partial = 0.f;
#pragma unroll
    for (int j = 0; j < 4; ++j) {
        const int l = tid + j * 256;
        float acc = 0.f;
#pragma unroll
        for (int h = 0; h < H_; ++h)
            acc += es[h * hstride + l] * et[h * hstride + l];
        acc *= (1.0f / H_);
        vals[j] = acc;
        partial += acc;
    }

    __shared__ float red[256];
    red[tid] = partial;
    __syncthreads();
    for (int off = 128; off > 0; off >>= 1) {
        if (tid < off) red[tid] += red[tid + off];
        __syncthreads();
    }
    const float inv = 1.0f / (red[0] + 1e-5f);

    float* out = ht + (size_t)bid * L_;
#pragma unroll
    for (int j = 0; j < 4; ++j)
        out[tid + j * 256] = vals[j] * inv;
}

// ---------------------------------------------------------------------------
// Kernel 3: feature_map[b, r, d] = sum_l ht[b, r, l] * seq[b, l, d]
// Per batch: (1764 x 1024) x (1024 x 768) fp32 GEMM.
// Block = 256 threads (8 waves), block tile 16M x 128N, K panels of 64,
// double-buffered async global->LDS staging; each wave computes one 16x16
// tile via V_WMMA_F32_16X16X4_F32 fed from LDS (ds_load_b64 / ds_load_b32).
// ---------------------------------------------------------------------------
__global__ void __launch_bounds__(256)
feature_gemm_kernel(const float* __restrict__ ht,
                    const float* __restrict__ seq,
                    float*       __restrict__ out) {
    // A panel: 16 rows x KB_ k, row stride 68 floats (bank-conflict-free frags)
    // B panel: KB_ rows x 128 cols, row stride 132 floats
    __shared__ __align__(16) float sA[2][16 * AS_];
    __shared__ __align__(16) float sB[2][KB_ * BS_];

    const int tid  = threadIdx.x;
    const int wave = tid >> 5;
    const int lane = tid & 31;
    const int half = lane >> 4;       // 0 or 1
    const int lm   = lane & 15;       // A: row-in-tile, B: col-in-tile

    const int b       = blockIdx.z;
    const int row0    = blockIdx.y * 16;        // 0..1760 (111 tiles)
    const int colBase = blockIdx.x * BN_;       // 0..640  (6 blocks)
    const int colw    = wave * 16;              // wave's column sub-tile

    const float* ht_b  = ht  + (size_t)b * NR_ * L_;
    const float* seq_b = seq + (size_t)b * L_ * D_;
    float*       out_b = out + (size_t)b * NR_ * D_;

    // ---- async-load address setup --------------------------------------
    // A: thread t loads 4 floats: row = t>>4 (0..15), kgroup = t&15
    const int a_row = tid >> 4;
    const int a_kg  = tid & 15;
    int ga_row = row0 + a_row;
    if (ga_row >= NR_) ga_row = NR_ - 1;        // clamp: garbage rows never stored
    const float* a_gbase = ht_b + (size_t)ga_row * L_ + a_kg * 4;
    uint32_t a_lds[2] = { lds_off(&sA[0][a_row * AS_ + a_kg * 4]),
                          lds_off(&sA[1][a_row * AS_ + a_kg * 4]) };

    // B: thread t loads 8 chunks of 4 floats: colgroup = t&31, rows t>>5 + 8*i
    const int b_cg = tid & 31;
    const int b_r0 = tid >> 5;
    const float* b_gbase = seq_b + (size_t)b_r0 * D_ + colBase + b_cg * 4;
    uint32_t b_lds[2] = { lds_off(&sB[0][b_r0 * BS_ + b_cg * 4]),
                          lds_off(&sB[1][b_r0 * BS_ + b_cg * 4]) };

    auto load_panel = [&](int k0, int buf) {
        async_load_b128(a_lds[buf], a_gbase + k0);
        const float* bg = b_gbase + (size_t)k0 * D_;
#pragma unroll
        for (int i = 0; i < 8; ++i)
            async_load_b128(b_lds[buf] + (uint32_t)(i * 8 * BS_ * 4),
                            bg + (size_t)i * 8 * D_);
    };

    // ---- pipeline: prologue -------------------------------------------
    load_panel(0, 0);
    wait_async0();
    __syncthreads();

    v8f acc = {};

    for (int p = 0; p < NPAN_; ++p) {
        const int cur = p & 1;
        if (p + 1 < NPAN_)
            load_panel((p + 1) * KB_, cur ^ 1);   // async prefetch next panel

        // compute on current panel: 16 K-steps of 4
#pragma unroll
        for (int ks = 0; ks < KB_ / 4; ++ks) {
            const int kl = ks * 4;

            // A fragment: ds_load_b64 (lane {half,lm} -> A[lm][kl+2*half .. +1])
            const v2f a = *(const v2f*)&sA[cur][lm * AS_ + kl + 2 * half];

            // B fragment: two ds_load_b32 (rows kl+2*half, kl+2*half+1)
            v2f bm;
            bm[0] = sB[cur][(kl + 2 * half)     * BS_ + colw + lm];
            bm[1] = sB[cur][(kl + 2 * half + 1) * BS_ + colw + lm];

            acc = __builtin_amdgcn_wmma_f32_16x16x4_f32(
                /*neg_a=*/false, a, /*neg_b=*/false, bm,
                /*c_mod=*/(short)0, acc, /*reuse_a=*/false, /*reuse_b=*/false);
        }

        wait_async0();      // next panel's async loads have landed in LDS
        __syncthreads();
    }

    // ---- store with row guard (1764 % 16 != 0) -------------------------
#pragma unroll
    for (int i = 0; i < 8; ++i) {
        const int m = i + 8 * half;
        const int r = row0 + m;
        if (r < NR_)
            out_b[(size_t)r * D_ + colBase + colw + lm] = acc[i];
    }
}

// ---------------------------------------------------------------------------
extern "C" void kernel_launch(void* const* d_in, const int* in_sizes, int n_in,
                              void* d_out, int out_size, void* d_ws, size_t ws_size,
                              hipStream_t stream) {
    const float* seq          = (const float*)d_in[0];  // [B,L,D]
    const float* attention    = (const float*)d_in[1];  // [B,H,L,L]
    const int*   mention_idx  = (const int*)d_in[2];    // [B,NE,M]
    const int*   mention_mask = (const int*)d_in[3];    // [B,NE,M]
    float*       out          = (float*)d_out;          // [B,NE,NE,D]

    // Workspace layout (fp32): ent_att [B,H,NE,L] | ht [B,NR,L]  (~37 MB)
    float* ent_att = (float*)d_ws;
    float* ht      = ent_att + (size_t)B_ * H_ * NE_ * L_;

    ent_att_kernel<<<dim3(B_ * H_ * NE_), dim3(256), 0, stream>>>(
        attention, mention_idx, mention_mask, ent_att);

    ht_att_kernel<<<dim3(B_ * NR_), dim3(256), 0, stream>>>(ent_att, ht);

    // grid: x = 768/128 col blocks, y = ceil(1764/16) row tiles, z = batch
    feature_gemm_kernel<<<dim3(D_ / BN_, (NR_ + 15) / 16, B_), dim3(256), 0, stream>>>(
        ht, seq, out);
}